// AssembleAttentionBlock_32169305047595
// MI455X (gfx1250) — compile-verified
//
#include <hip/hip_runtime.h>
#include <hip/hip_bf16.h>

// Problem constants (match reference)
#define DIMN   3072
#define NHEADS 24
#define HDIM   128
#define BB     2
#define NIMG   4096
#define RR     8
#define TT     512
#define MROWS  (BB * RR)   // 16 — exactly one WMMA M-tile

typedef __attribute__((ext_vector_type(2))) float v2f;
typedef __attribute__((ext_vector_type(4))) float v4f;
typedef __attribute__((ext_vector_type(8))) float v8f;

// -----------------------------------------------------------------------------
// C(16 x DIMN) = A(16 x DIMN) @ W^T  [+ bias, * (1-alpha[m]) when SCALED]
//   C[m,n] = sum_k A[m,k] * W[n,k]
// One wave per 16-column tile, full-FP32 WMMA accumulation.
//
// V_WMMA_F32_16X16X4_F32 operand layout (ISA 7.12.2):
//   A (16x4):  lanes 0-15 -> M=lane, VGPR0/1 = slots K0/K1
//              lanes 16-31-> M=lane-16, VGPR0/1 = slots K2/K3
//   B (4x16):  lanes 0-15 -> N=lane, VGPR0/1 = K0/K1 ; lanes 16-31 -> K2/K3
//   C/D:       VGPR v, lane l -> M = v + 8*(l>=16), N = l%16
//
// K remap for b128 loads: per 16-K iteration, WMMA t (t=0..3) covers global
//   half0 slots = {k+2t, k+2t+1},  half1 slots = {k+8+2t, k+8+2t+1}
// so each lane streams contiguous float4s at (row + k + half*8).
// -----------------------------------------------------------------------------
template <bool SCALED>
__global__ __launch_bounds__(32)
void gemm16_wmma_f32(const float* __restrict__ A,      // 16 x DIMN, row-major
                     const float* __restrict__ W,      // DIMN x DIMN, row-major (used transposed)
                     const float* __restrict__ bias,   // DIMN      (SCALED only)
                     const float* __restrict__ alpha,  // 16 flat   (SCALED only)
                     float* __restrict__ C)            // 16 x DIMN, row-major
{
    const int n0   = blockIdx.x * 16;
    const int lane = threadIdx.x & 31;
    const int half = lane >> 4;
    const int lm   = lane & 15;

    const float* arow = A + (size_t)lm * DIMN + half * 8;          // A row M = lm
    const float* wrow = W + (size_t)(n0 + lm) * DIMN + half * 8;   // W row n -> B col n

    v8f acc0 = {}, acc1 = {};
#pragma unroll 2
    for (int k = 0; k < DIMN; k += 16) {
        v4f aLo = *(const v4f*)(arow + k);
        v4f aHi = *(const v4f*)(arow + k + 4);
        v4f bLo = *(const v4f*)(wrow + k);
        v4f bHi = *(const v4f*)(wrow + k + 4);

        v2f a, b;
        a.x = aLo.x; a.y = aLo.y; b.x = bLo.x; b.y = bLo.y;
        acc0 = __builtin_amdgcn_wmma_f32_16x16x4_f32(false, a, false, b,
                                                     (short)0, acc0, false, false);
        a.x = aLo.z; a.y = aLo.w; b.x = bLo.z; b.y = bLo.w;
        acc1 = __builtin_amdgcn_wmma_f32_16x16x4_f32(false, a, false, b,
                                                     (short)0, acc1, false, false);
        a.x = aHi.x; a.y = aHi.y; b.x = bHi.x; b.y = bHi.y;
        acc0 = __builtin_amdgcn_wmma_f32_16x16x4_f32(false, a, false, b,
                                                     (short)0, acc0, false, false);
        a.x = aHi.z; a.y = aHi.w; b.x = bHi.z; b.y = bHi.w;
        acc1 = __builtin_amdgcn_wmma_f32_16x16x4_f32(false, a, false, b,
                                                     (short)0, acc1, false, false);
    }

    float r[8];
#pragma unroll
    for (int v = 0; v < 8; ++v) r[v] = acc0[v] + acc1[v];

    if (SCALED) {
        const float bval = bias[n0 + lm];
        // rows handled by this lane: m = half*8 + v, v = 0..7 (contiguous)
        v4f al0 = *(const v4f*)(alpha + half * 8);
        v4f al1 = *(const v4f*)(alpha + half * 8 + 4);
        float sc[8] = {1.0f - al0.x, 1.0f - al0.y, 1.0f - al0.z, 1.0f - al0.w,
                       1.0f - al1.x, 1.0f - al1.y, 1.0f - al1.z, 1.0f - al1.w};
#pragma unroll
        for (int v = 0; v < 8; ++v) r[v] = (r[v] + bval) * sc[v];
    }

    float* crow = C + (size_t)(half * 8) * DIMN + n0 + lm;
#pragma unroll
    for (int v = 0; v < 8; ++v)
        crow[(size_t)v * DIMN] = r[v];
}

// -----------------------------------------------------------------------------
// Scatter-broadcast. softmax over a singleton key axis == 1, so the attention
// output row for every token in ref r is exactly y[b*R + r, :]. img_idxs is a
// disjoint partition of all NIMG tokens (R*T == NIMG), so the whole output is
// overwritten and image_tokens never reaches the output. One 256-thread block
// per output token row, float4 stores; the 16 source rows stay L2-resident.
// -----------------------------------------------------------------------------
__global__ __launch_bounds__(256)
void bcast_scatter(const float* __restrict__ y,       // 16 x DIMN
                   const int*   __restrict__ idx,     // R*T flat indices
                   float* __restrict__ out)           // B x NIMG x DIMN
{
    const int j = blockIdx.x & (NIMG - 1);   // position within (R*T)
    const int b = blockIdx.x >> 12;          // NIMG = 4096 = 2^12
    const int dstn = idx[j];
    const int m = b * RR + (j >> 9);         // T = 512 = 2^9

    const float4* src = (const float4*)(y + (size_t)m * DIMN);
    float4* dst = (float4*)(out + ((size_t)b * NIMG + (size_t)dstn) * DIMN);
#pragma unroll
    for (int t = threadIdx.x; t < DIMN / 4; t += 256)
        dst[t] = src[t];
}

extern "C" void kernel_launch(void* const* d_in, const int* in_sizes, int n_in,
                              void* d_out, int out_size, void* d_ws, size_t ws_size,
                              hipStream_t stream) {
    // setup_inputs() order:
    const float* instance_tokens = (const float*)d_in[0];  // (B,R,DIM)
    // d_in[1] image_tokens: dead (scatter fully overwrites output)
    // d_in[2] bbox_masks:   unused by forward
    const float* alpha    = (const float*)d_in[3];         // (B,R)
    const float* W_layout = (const float*)d_in[4];         // (DIM,DIM)
    const float* b_layout = (const float*)d_in[5];         // (DIM,)
    // d_in[6] W_q: dead (softmax over singleton axis == 1)
    const float* W_out    = (const float*)d_in[7];         // (DIM,DIM)
    const int*   img_idxs = (const int*)d_in[8];           // (R,T)

    float* out = (float*)d_out;

    float* kv = (float*)d_ws;                  // 16 x DIMN scaled layout_kv
    float* y  = kv + (size_t)MROWS * DIMN;     // 16 x DIMN projected rows
    (void)ws_size; (void)out_size; (void)n_in; (void)in_sizes;

    dim3 gGemm(DIMN / 16), bGemm(32);
    // kv = (instance_tokens @ W_layout^T + b_layout) * (1 - alpha)
    gemm16_wmma_f32<true><<<gGemm, bGemm, 0, stream>>>(instance_tokens, W_layout,
                                                       b_layout, alpha, kv);
    // y = kv @ W_out^T
    gemm16_wmma_f32<false><<<gGemm, bGemm, 0, stream>>>(kv, W_out, nullptr, nullptr, y);
    // out[b, idx[j], :] = y[b*R + j/T, :]
    bcast_scatter<<<dim3(BB * NIMG), dim3(256), 0, stream>>>(y, img_idxs, out);
}